// MultiHeadAttention_51969104281902
// MI455X (gfx1250) — compile-verified
//
#include <hip/hip_runtime.h>
#include <hip/hip_bf16.h>

typedef float v2f __attribute__((ext_vector_type(2)));
typedef float v8f __attribute__((ext_vector_type(8)));

#define NEG_INF (-__builtin_huge_valf())

static constexpr int cNH = 8;
static constexpr int cB  = 64;
static constexpr int cGS = 201;
static constexpr int cD  = 128;
static constexpr int cKD = 16;
static constexpr int cNP = 100;          // npick
static constexpr float cNF = 0.25f;      // 1/sqrt(16)
static constexpr int SST = 420;          // S row stride (floats), bank-conflict-free
static constexpr int KVS = 18;           // K/V LDS row stride (floats)
static constexpr int KVP = 216;          // K/V LDS rows (zero padded past 201; max idx used = 212)
static constexpr int QSS = 18;           // Q-stage row stride
static constexpr int WLS = 132;          // transposed weight-tile stride (conflict-free b64 reads)

// D(16x16,f32) = A(16x4,f32) * B(4x16,f32) + C   -- native fp32 WMMA
__device__ __forceinline__ v8f wmma_f32_k4(v2f a, v2f b, v8f c) {
  return __builtin_amdgcn_wmma_f32_16x16x4_f32(false, a, false, b, (short)0, c, false, false);
}

// ---------------------------------------------------------------------------
// Kernel 1: projections  X(rows x 128) @ W[h](128 x 16)  for 9 jobs
// grid (B, NH, 9), block 128 (4 waves, 1 wave per 16-row tile)
// ---------------------------------------------------------------------------
__global__ __launch_bounds__(128)
void k1_proj(const float* __restrict__ q,
             const float* __restrict__ Wq, const float* __restrict__ Wk, const float* __restrict__ Wv,
             const float* __restrict__ W1, const float* __restrict__ W2, const float* __restrict__ W3,
             const float* __restrict__ W4, const float* __restrict__ W5, const float* __restrict__ W6,
             float* __restrict__ oQ, float* __restrict__ oK, float* __restrict__ oV,
             float* __restrict__ o1, float* __restrict__ o2, float* __restrict__ o3,
             float* __restrict__ o4, float* __restrict__ o5, float* __restrict__ o6)
{
  const int b = blockIdx.x, h = blockIdx.y, z = blockIdx.z;
  const float* Wj; float* Oj; int rowoff, nrows;
  switch (z) {
    case 0: Wj = Wq; Oj = oQ; rowoff = 0;   nrows = cGS; break;
    case 1: Wj = Wk; Oj = oK; rowoff = 0;   nrows = cGS; break;
    case 2: Wj = Wv; Oj = oV; rowoff = 0;   nrows = cGS; break;
    case 3: Wj = W1; Oj = o1; rowoff = 1;   nrows = cNP; break;
    case 4: Wj = W2; Oj = o2; rowoff = 1;   nrows = cNP; break;
    case 5: Wj = W3; Oj = o3; rowoff = 1;   nrows = cNP; break;
    case 6: Wj = W4; Oj = o4; rowoff = 101; nrows = cNP; break;
    case 7: Wj = W5; Oj = o5; rowoff = 101; nrows = cNP; break;
    default:Wj = W6; Oj = o6; rowoff = 101; nrows = cNP; break;
  }

  __shared__ float Wlt[cKD * WLS];   // transposed: Wlt[n*WLS + k] = W[k][n]
  for (int i = threadIdx.x; i < cD * cKD; i += 128) {
    const int k = i >> 4, n = i & 15;
    Wlt[n * WLS + k] = Wj[(size_t)h * cD * cKD + i];
  }
  __syncthreads();

  const int wid = threadIdx.x >> 5, lane = threadIdx.x & 31;
  const int fc  = lane & 15;          // fragment M-row (A) / N-col (B,D)
  const int kk  = (lane >> 4) * 2;    // K sub-offset within 4-chunk
  const int hs8 = (lane >> 4) * 8;    // C/D row offset for upper half-wave
  const int ntiles = (nrows + 15) >> 4;
  const float* qb = q + (size_t)b * cGS * cD;
  float* Ob = Oj + (size_t)(h * cB + b) * nrows * cKD;

  for (int t = wid; t < ntiles; t += 4) {
    const int rbase = t * 16;
    const int arow  = rbase + fc;
    const int crow  = (arow < nrows) ? arow : (nrows - 1);   // clamp: dead rows never stored
    const float* arp = qb + (size_t)(rowoff + crow) * cD + kk;
    // issue all A loads as a clause, then the WMMA chain against LDS B
    v2f areg[32];
    #pragma unroll
    for (int k4 = 0; k4 < 32; ++k4)
      areg[k4] = *(const v2f*)(arp + 4 * k4);
    v8f acc = {};
    #pragma unroll
    for (int k4 = 0; k4 < 32; ++k4) {
      const v2f bf = *(const v2f*)&Wlt[fc * WLS + 4 * k4 + kk];
      acc = wmma_f32_k4(areg[k4], bf, acc);
    }
    #pragma unroll
    for (int v = 0; v < 8; ++v) {
      const int row = rbase + v + hs8;
      if (row < nrows) Ob[(size_t)row * cKD + fc] = acc[v];
    }
  }
}

// ---------------------------------------------------------------------------
// Kernel 2: fused logits + softmax + mixing per (row-tile, b, h)
// grid (13, B, NH), block 256 (8 waves)
// Uniform per-row logit layout in LDS row of width 416 (stride 420):
//   [0..203]   base  (201 valid, 3 pad)
//   [204]      diag  (205..207 pad)
//   [208..311] extraA (100 valid, 4 pad)
//   [312..415] extraB (100 valid, 4 pad)
// ---------------------------------------------------------------------------
__global__ __launch_bounds__(256)
void k2_attn(const float* __restrict__ Qh, const float* __restrict__ Kh, const float* __restrict__ Vh,
             const float* __restrict__ Q1, const float* __restrict__ Q2, const float* __restrict__ Q3,
             const float* __restrict__ Q4, const float* __restrict__ Q5, const float* __restrict__ Q6,
             float* __restrict__ heads)
{
  const int rt = blockIdx.x, b = blockIdx.y, h = blockIdx.z;
  const int r0 = rt * 16;
  const int hb = h * cB + b;
  const int tid = threadIdx.x;
  const int wid = tid >> 5, lane = tid & 31;
  const int fc  = lane & 15;
  const int kk  = (lane >> 4) * 2;
  const int hs8 = (lane >> 4) * 8;

  __shared__ float Ks[KVP * KVS];      // 15552 B
  __shared__ float Vs[KVP * KVS];      // 15552 B
  __shared__ float Sm[16 * SST];       // 26880 B : logits -> probabilities
  __shared__ float Qst[16 * QSS], QAst[16 * QSS], QBst[16 * QSS], Qdst[16 * QSS];
  __shared__ int   didx[16];
  __shared__ float Os[256];            // 16x16 output accumulator

  // ---- phase 0: stage K, V, per-row Q fragments ----
  const float* Kb = Kh + (size_t)hb * cGS * cKD;
  const float* Vb = Vh + (size_t)hb * cGS * cKD;
  for (int i = tid; i < KVP * cKD; i += 256) {
    const int row = i >> 4, k = i & 15;
    const bool v  = row < cGS;
    Ks[row * KVS + k] = v ? Kb[i] : 0.f;
    Vs[row * KVS + k] = v ? Vb[i] : 0.f;
  }
  {
    const int m = tid >> 4, k = tid & 15;
    const int r = r0 + m;
    float qv = 0.f, qa = 0.f, qb2 = 0.f, qd = 0.f;
    if (r < cGS) {
      qv = Qh[((size_t)hb * cGS + r) * cKD + k];
      if (r >= 1 && r <= 100) {                 // pick row i = r-1
        const size_t base = ((size_t)hb * cNP + (r - 1)) * cKD + k;
        qa = Q2[base]; qb2 = Q3[base]; qd = Q1[base];
      } else if (r >= 101 && r <= 200) {        // deliv row j = r-101
        const size_t base = ((size_t)hb * cNP + (r - 101)) * cKD + k;
        qa = Q5[base]; qb2 = Q6[base]; qd = Q4[base];
      }
    }
    Qst[m * QSS + k] = qv; QAst[m * QSS + k] = qa;
    QBst[m * QSS + k] = qb2; Qdst[m * QSS + k] = qd;
    if (k == 0)
      didx[m] = (r >= 1 && r <= 100) ? (100 + r) : ((r >= 101 && r <= 200) ? (r - 100) : 0);
  }
  Os[tid] = 0.f;
  __syncthreads();

  // ---- phase 1: logits via WMMA. 27 column-tile jobs across 8 waves ----
  for (int jj = wid; jj < 27; jj += 8) {
    if (jj < 13) {                               // base block: Q @ K^T, cols jj*16..
      const int j0 = jj * 16;
      v8f acc = {};
      #pragma unroll
      for (int k0 = 0; k0 < cKD; k0 += 4) {
        const v2f a  = *(const v2f*)&Qst[fc * QSS + k0 + kk];
        const v2f bf = *(const v2f*)&Ks[(j0 + fc) * KVS + k0 + kk];
        acc = wmma_f32_k4(a, bf, acc);
      }
      const int col = j0 + fc;
      #pragma unroll
      for (int v = 0; v < 8; ++v) {
        const int m = v + hs8, r = r0 + m;
        const float val = (col < cGS && r < cGS) ? cNF * acc[v] : NEG_INF;
        if (col < 204) Sm[m * SST + col] = val;
      }
    } else {                                     // extraA / extraB blocks
      const bool isA = (jj < 20);
      const int  c0  = (isA ? (jj - 13) : (jj - 20)) * 16;
      const float* Qx = isA ? QAst : QBst;
      // pick-variant B rows: extraA->K_pk(1+), extraB->K_del(101+); deliv is swapped
      const int rp = (isA ? 1 : 101) + c0 + fc;
      const int rd = (isA ? 101 : 1) + c0 + fc;
      v8f ap = {}, ad = {};
      #pragma unroll
      for (int k0 = 0; k0 < cKD; k0 += 4) {
        const v2f a  = *(const v2f*)&Qx[fc * QSS + k0 + kk];
        const v2f b1 = *(const v2f*)&Ks[rp * KVS + k0 + kk];
        const v2f b2 = *(const v2f*)&Ks[rd * KVS + k0 + kk];
        ap = wmma_f32_k4(a, b1, ap);
        ad = wmma_f32_k4(a, b2, ad);
      }
      const int cloc = c0 + fc;
      const int colbase = isA ? 208 : 312;
      #pragma unroll
      for (int v = 0; v < 8; ++v) {
        const int m = v + hs8, r = r0 + m;
        const bool ok = (cloc < cNP) && (r >= 1) && (r <= 200);
        const float val = ok ? cNF * ((r <= 100) ? ap[v] : ad[v]) : NEG_INF;
        if (cloc < 104) Sm[m * SST + colbase + cloc] = val;
      }
    }
  }
  // diagonal terms (col 204) + pad cols 205..207
  if (wid == 0 && lane < 16) {
    const int m = lane, r = r0 + m;
    const int di = didx[m];
    float s = 0.f;
    #pragma unroll
    for (int k = 0; k < cKD; ++k) s += Qdst[m * QSS + k] * Ks[di * KVS + k];
    const bool dv = (r >= 1) && (r <= 200);
    Sm[m * SST + 204] = dv ? cNF * s : NEG_INF;
    Sm[m * SST + 205] = NEG_INF;
    Sm[m * SST + 206] = NEG_INF;
    Sm[m * SST + 207] = NEG_INF;
  }
  __syncthreads();

  // ---- phase 2: softmax (one wave per 2 rows, shuffle reductions) ----
  for (int m = wid * 2; m < wid * 2 + 2; ++m) {
    const int r = r0 + m;
    if (r >= cGS) continue;
    float* Sr = &Sm[m * SST];
    float mx = NEG_INF;
    for (int c = lane; c < 416; c += 32) mx = fmaxf(mx, Sr[c]);
    for (int o = 16; o > 0; o >>= 1) mx = fmaxf(mx, __shfl_xor(mx, o, 32));
    float sm = 0.f;
    for (int c = lane; c < 416; c += 32) {
      const float e = expf(Sr[c] - mx);   // exp(-inf - mx) == 0 handles all masks
      Sr[c] = e; sm += e;
    }
    for (int o = 16; o > 0; o >>= 1) sm += __shfl_xor(sm, o, 32);
    const float inv = 1.f / sm;
    for (int c = lane; c < 416; c += 32) Sr[c] *= inv;
  }
  __syncthreads();

  // ---- phase 3: mixing O = P_base@V + class-select(P_A,P_B @ V_pk/V_del) ----
  v8f ab = {}, a1 = {}, a2 = {}, a3 = {}, a4 = {};
  for (int jid = wid; jid < 155; jid += 8) {
    int pcol, vrow, tgt;
    if (jid < 51)       { tgt = 0; pcol = jid * 4;                 vrow = jid * 4; }
    else if (jid < 77)  { tgt = 1; int kc = jid - 51;  pcol = 208 + 4 * kc; vrow = 1   + 4 * kc; }
    else if (jid < 103) { tgt = 2; int kc = jid - 77;  pcol = 208 + 4 * kc; vrow = 101 + 4 * kc; }
    else if (jid < 129) { tgt = 3; int kc = jid - 103; pcol = 312 + 4 * kc; vrow = 101 + 4 * kc; }
    else                { tgt = 4; int kc = jid - 129; pcol = 312 + 4 * kc; vrow = 1   + 4 * kc; }
    v2f a, bf;
    a = *(const v2f*)&Sm[fc * SST + pcol + kk];
    bf.x = Vs[(vrow + kk) * KVS + fc];
    bf.y = Vs[(vrow + kk + 1) * KVS + fc];
    v8f* tp = (tgt == 0) ? &ab : (tgt == 1) ? &a1 : (tgt == 2) ? &a2 : (tgt == 3) ? &a3 : &a4;
    *tp = wmma_f32_k4(a, bf, *tp);
  }
  #pragma unroll
  for (int v = 0; v < 8; ++v) {
    const int m = v + hs8, r = r0 + m;
    float extra = 0.f;
    if (r >= 1 && r <= 100)        extra = a1[v] + a3[v];   // pick: P_A@V_pk + P_B@V_del
    else if (r >= 101 && r <= 200) extra = a2[v] + a4[v];   // deliv: P_A@V_del + P_B@V_pk
    atomicAdd(&Os[m * 16 + fc], ab[v] + extra);
  }
  __syncthreads();

  // ---- phase 4: add diag value contribution, write heads ----
  {
    const int m = tid >> 4, n = tid & 15, r = r0 + m;
    if (r < cGS) {
      const float o = Os[m * 16 + n] + Sm[m * SST + 204] * Vs[didx[m] * KVS + n];
      heads[((size_t)b * cGS + r) * cD + h * cKD + n] = o;
    }
  }
}

// ---------------------------------------------------------------------------
// Kernel 3: out = heads(12864 x 128) @ Wout(128 x 128)
// grid (201, 8), block 128 (4 waves, 1 wave per exact 16-row tile)
// ---------------------------------------------------------------------------
__global__ __launch_bounds__(128)
void k3_out(const float* __restrict__ heads, const float* __restrict__ Wout,
            float* __restrict__ out)
{
  const int nt = blockIdx.y;
  __shared__ float Wlt[cKD * WLS];   // transposed N-tile of Wout: Wlt[n*WLS + f]
  for (int i = threadIdx.x; i < cD * cKD; i += 128) {
    const int f = i >> 4, n = i & 15;
    Wlt[n * WLS + f] = Wout[(size_t)f * cD + nt * 16 + n];
  }
  __syncthreads();

  const int wid = threadIdx.x >> 5, lane = threadIdx.x & 31;
  const int fc = lane & 15, kk = (lane >> 4) * 2, hs8 = (lane >> 4) * 8;
  const int mt = blockIdx.x * 4 + wid;        // 0..803 (exact: 12864 = 804*16)
  const size_t rbase = (size_t)mt * 16;
  const float* ar = heads + (rbase + fc) * cD + kk;

  v2f areg[32];
  #pragma unroll
  for (int k4 = 0; k4 < 32; ++k4)
    areg[k4] = *(const v2f*)(ar + 4 * k4);
  v8f acc = {};
  #pragma unroll
  for (int k4 = 0; k4 < 32; ++k4) {
    const v2f bf = *(const v2f*)&Wlt[fc * WLS + 4 * k4 + kk];
    acc = wmma_f32_k4(areg[k4], bf, acc);
  }
  #pragma unroll
  for (int v = 0; v < 8; ++v) {
    const size_t row = rbase + v + hs8;
    out[row * cD + nt * 16 + fc] = acc[v];
  }
}

// ---------------------------------------------------------------------------
extern "C" void kernel_launch(void* const* d_in, const int* in_sizes, int n_in,
                              void* d_out, int out_size, void* d_ws, size_t ws_size,
                              hipStream_t stream) {
  (void)in_sizes; (void)n_in; (void)out_size; (void)ws_size;
  const float* q  = (const float*)d_in[0];
  const float* Wq = (const float*)d_in[1];
  const float* Wk = (const float*)d_in[2];
  const float* Wv = (const float*)d_in[3];
  const float* W1 = (const float*)d_in[4];
  const float* W2 = (const float*)d_in[5];
  const float* W3 = (const float*)d_in[6];
  const float* W4 = (const float*)d_in[7];
  const float* W5 = (const float*)d_in[8];
  const float* W6 = (const float*)d_in[9];
  const float* Wo = (const float*)d_in[10];
  float* out = (float*)d_out;

  // workspace partition (floats): 3 full projections + 6 pick/deliv + heads
  float* ws = (float*)d_ws;
  const size_t szFull = (size_t)cNH * cB * cGS * cKD;   // 1,646,592
  const size_t szPick = (size_t)cNH * cB * cNP * cKD;   //   819,200
  float* Qh = ws; ws += szFull;
  float* Kh = ws; ws += szFull;
  float* Vh = ws; ws += szFull;
  float* Q1 = ws; ws += szPick;
  float* Q2 = ws; ws += szPick;
  float* Q3 = ws; ws += szPick;
  float* Q4 = ws; ws += szPick;
  float* Q5 = ws; ws += szPick;
  float* Q6 = ws; ws += szPick;
  float* heads = ws;                                     // 1,646,592

  k1_proj<<<dim3(cB, cNH, 9), 128, 0, stream>>>(q, Wq, Wk, Wv, W1, W2, W3, W4, W5, W6,
                                                Qh, Kh, Vh, Q1, Q2, Q3, Q4, Q5, Q6);
  k2_attn<<<dim3(13, cB, cNH), 256, 0, stream>>>(Qh, Kh, Vh, Q1, Q2, Q3, Q4, Q5, Q6, heads);
  k3_out<<<dim3(cGS, cNH), 128, 0, stream>>>(heads, Wo, out);
}